// LossManager_27161373180439
// MI455X (gfx1250) — compile-verified
//
#include <hip/hip_runtime.h>
#include <hip/hip_bf16.h>
#include <math.h>

#define NUM_CLASSES 19
#define IMG_B 4
#define IMG_HW 524288            // 512*1024 = 2^19
#define NPIX (IMG_B * IMG_HW)    // 2097152
#define NBINS 8192
#define BIN_SHIFT 19             // 32 - log2(NBINS)
#define NBLK1 1024
#define THR1 256
#define ITER1 (NPIX / (NBLK1 * THR1))   // exactly 8, no tail

// workspace layout, 32-bit units
#define WS_CE     0
#define WS_FOCAL  1
#define WS_INTER  4                         // 19 f32
#define WS_COUNT  32                        // 19 u32
#define WS_PSUM   64                        // 19 f32 (WMMA-reduced class prob sums)
#define WS_LOV    96                        // 19 f32
#define WS_PART   128                       // NBLK1*32 f32 (per-block class partials)
#define WS_HIST   (WS_PART + NBLK1 * 32)    // 19*2*NBINS u32 (bg plane, fg plane)
#define WS_TOTAL_U32 (WS_HIST + NUM_CLASSES * 2 * NBINS)

typedef __attribute__((ext_vector_type(2))) float v2f;
typedef __attribute__((ext_vector_type(8))) float v8f;

#if defined(__has_builtin)
#if __has_builtin(__builtin_amdgcn_global_load_async_to_lds_b32) && \
    __has_builtin(__builtin_amdgcn_s_wait_asynccnt)
#define HAVE_ASYNC_LDS 1
#endif
#endif

#ifdef HAVE_ASYNC_LDS
typedef __attribute__((address_space(1))) int gas_int;   // prints as '__device__ int'
typedef __attribute__((address_space(3))) int las_int;   // LDS
#endif

__global__ void k_zero(unsigned* ws) {
    for (unsigned i = blockIdx.x * blockDim.x + threadIdx.x; i < WS_TOTAL_U32;
         i += gridDim.x * blockDim.x)
        ws[i] = 0u;
}

// Pass 1: streaming softmax / CE / focal / dice partials / lovasz histograms.
// Mask stream is double-buffered into LDS with async global->LDS loads
// (GLOBAL_LOAD_ASYNC_TO_LDS_B32 + s_wait_asynccnt) overlapped with softmax math.
__global__ __launch_bounds__(THR1) void k_main(const float* __restrict__ lgts,
                                               const int* __restrict__ masks,
                                               float* __restrict__ wsf,
                                               unsigned* __restrict__ wsu) {
    __shared__ float s_inter[NUM_CLASSES];
    __shared__ float s_psum[NUM_CLASSES];
    __shared__ unsigned s_cnt[NUM_CLASSES];
    __shared__ float s_cf[2];
#ifdef HAVE_ASYNC_LDS
    __shared__ int s_mask[2][THR1];
#endif

    int tid = threadIdx.x;
    if (tid < NUM_CLASSES) { s_inter[tid] = 0.f; s_psum[tid] = 0.f; s_cnt[tid] = 0u; }
    if (tid < 2) s_cf[tid] = 0.f;
    __syncthreads();

    unsigned* hist = wsu + WS_HIST;
    float ce_acc = 0.f, fo_acc = 0.f;
    float ps[NUM_CLASSES];
#pragma unroll
    for (int c = 0; c < NUM_CLASSES; ++c) ps[c] = 0.f;

    const int stride = NBLK1 * THR1;
    const int p0 = blockIdx.x * THR1 + tid;

#ifdef HAVE_ASYNC_LDS
    // prime the pipeline: stage tile 0 of the mask stream
    __builtin_amdgcn_global_load_async_to_lds_b32(
        (gas_int*)(masks + p0), (las_int*)&s_mask[0][tid], 0, 0);
#endif

    for (int it = 0; it < ITER1; ++it) {
        int p = p0 + it * stride;
        int img = p >> 19;                 // p / IMG_HW
        int pix = p & (IMG_HW - 1);
        long long base = (long long)img * (NUM_CLASSES * (long long)IMG_HW) + pix;

        int t;
#ifdef HAVE_ASYNC_LDS
        if (it + 1 < ITER1) {
            int pn = p + stride;
            __builtin_amdgcn_global_load_async_to_lds_b32(
                (gas_int*)(masks + pn), (las_int*)&s_mask[(it + 1) & 1][tid], 0, 0);
            __builtin_amdgcn_s_wait_asynccnt(1);   // in-order: current tile has landed
        } else {
            __builtin_amdgcn_s_wait_asynccnt(0);   // drain
        }
        t = s_mask[it & 1][tid];
#else
        t = masks[p];
#endif
        t = (t < 0) ? 0 : ((t >= NUM_CLASSES) ? NUM_CLASSES - 1 : t);

        // Prefetch next iteration's first logit cacheline (global_prefetch_b8)
        if (it + 1 < ITER1) {
            int pn = p + stride;
            const float* pf = lgts + (long long)(pn >> 19) * (NUM_CLASSES * (long long)IMG_HW)
                              + (pn & (IMG_HW - 1));
            __builtin_prefetch(pf, 0, 3);
        }

        float lg[NUM_CLASSES];
        float m = -3.4e38f;
#pragma unroll
        for (int c = 0; c < NUM_CLASSES; ++c) {
            lg[c] = lgts[base + (long long)c * IMG_HW];
            m = fmaxf(m, lg[c]);
        }
        float s = 0.f;
#pragma unroll
        for (int c = 0; c < NUM_CLASSES; ++c) s += __expf(lg[c] - m);
        float logsum = __logf(s) + m;

        float lgt = lg[0];
#pragma unroll
        for (int c = 1; c < NUM_CLASSES; ++c) lgt = (c == t) ? lg[c] : lgt;

        float logpt = lgt - logsum;
        ce_acc -= logpt;
        float pt = __expf(logpt);
        float omp = 1.f - pt;
        fo_acc += -0.25f * omp * omp * logpt;   // ALPHA=0.25, GAMMA=2

        atomicAdd(&s_inter[t], pt);             // ds_add_f32
        atomicAdd(&s_cnt[t], 1u);               // ds_add_u32

#pragma unroll
        for (int c = 0; c < NUM_CLASSES; ++c) {
            ps[c] += __expf(lg[c] - logsum);
            int fg = (c == t) ? 1 : 0;
            float err = fg ? (1.f - lg[c]) : (1.f + lg[c]);
            unsigned s32 = __float_as_uint(err);
            unsigned key = (s32 & 0x80000000u) ? ~s32 : (s32 | 0x80000000u); // order-preserving
            unsigned bin = key >> BIN_SHIFT;
            atomicAdd(&hist[(unsigned)(c * 2 + fg) * NBINS + bin], 1u);
        }
    }

    atomicAdd(&s_cf[0], ce_acc);
    atomicAdd(&s_cf[1], fo_acc);
#pragma unroll
    for (int c = 0; c < NUM_CLASSES; ++c) atomicAdd(&s_psum[c], ps[c]);
    __syncthreads();

    if (tid == 0) { atomicAdd(&wsf[WS_CE], s_cf[0]); atomicAdd(&wsf[WS_FOCAL], s_cf[1]); }
    if (tid < NUM_CLASSES) {
        atomicAdd(&wsf[WS_INTER + tid], s_inter[tid]);
        atomicAdd(&wsu[WS_COUNT + tid], s_cnt[tid]);
    }
    if (tid < 32)
        wsf[WS_PART + blockIdx.x * 32 + tid] = (tid < NUM_CLASSES) ? s_psum[tid] : 0.f;
}

// Pass 2: reduce NBLK1 x 32 partial matrix to 32 class totals with F32 WMMA.
// A = 16x4 all-ones, B = 4x16 partials tile; D[:,n] accumulates column sums.
__global__ void k_wmma_reduce(const float* __restrict__ part, float* __restrict__ psum_out) {
    int lane = threadIdx.x;          // one wave32, EXEC all ones
    int col  = lane & 15;
    int krow = (lane >> 4) * 2;      // lanes 0-15 -> K rows {0,1}; lanes 16-31 -> {2,3}
    v2f a; a.x = 1.0f; a.y = 1.0f;   // ones matrix (exact in f32)
    v8f c0 = {}; v8f c1 = {};
    for (int r = 0; r < NBLK1; r += 4) {
        v2f b0, b1;
        b0.x = part[(r + krow) * 32 + col];
        b0.y = part[(r + krow + 1) * 32 + col];
        b1.x = part[(r + krow) * 32 + 16 + col];
        b1.y = part[(r + krow + 1) * 32 + 16 + col];
        c0 = __builtin_amdgcn_wmma_f32_16x16x4_f32(false, a, false, b0, (short)0, c0, false, false);
        c1 = __builtin_amdgcn_wmma_f32_16x16x4_f32(false, a, false, b1, (short)0, c1, false, false);
    }
    // All D rows identical (A==ones); C/D VGPR0 on lanes 0-15 holds N=lane, M=0.
    float t0 = c0[0], t1 = c1[0];
    if (lane < 16) psum_out[lane] = t0;
    if (lane < NUM_CLASSES - 16) psum_out[16 + lane] = t1;
}

// Pass 3: per-class descending bin sweep -> telescoped Lovasz term.
__global__ void k_lovasz(const unsigned* __restrict__ wsu, float* __restrict__ lov_out) {
    int c = blockIdx.x;
    int lane = threadIdx.x;          // one wave32
    const unsigned* hbg = wsu + WS_HIST + (unsigned)(c * 2 + 0) * NBINS;
    const unsigned* hfg = wsu + WS_HIST + (unsigned)(c * 2 + 1) * NBINS;
    float P = (float)wsu[WS_COUNT + c];

    float runfg = 0.f, runbg = 0.f, acc = 0.f;
    for (int chunk = 0; chunk < NBINS; chunk += 32) {
        int bin = NBINS - 1 - (chunk + lane);   // descending error order
        float fg = (float)hfg[bin];
        float bg = (float)hbg[bin];
        float ifg = fg, ibg = bg;
        for (int off = 1; off < 32; off <<= 1) {  // inclusive wave32 scan
            float tf = __shfl_up(ifg, off, 32);
            float tb = __shfl_up(ibg, off, 32);
            if (lane >= off) { ifg += tf; ibg += tb; }
        }
        float cfg = runfg + ifg, cbg = runbg + ibg;
        float efg = cfg - fg,    ebg = cbg - bg;
        float Jb = 1.f - (P - efg) / (P + ebg);   // Jaccard before bin
        float Ja = 1.f - (P - cfg) / (P + cbg);   // after bin
        unsigned key = ((unsigned)bin << BIN_SHIFT) | (1u << (BIN_SHIFT - 1));
        unsigned s32 = (key & 0x80000000u) ? (key & 0x7fffffffu) : ~key;
        float err = __uint_as_float(s32);
        float rerr = err > 0.f ? err : 0.f;
        if (fg + bg > 0.f) acc += rerr * (Ja - Jb);
        runfg += __shfl(ifg, 31, 32);
        runbg += __shfl(ibg, 31, 32);
    }
    for (int off = 16; off > 0; off >>= 1) acc += __shfl_down(acc, off, 32);
    if (lane == 0) lov_out[c] = (P > 0.f) ? acc : 0.f;
}

__global__ void k_final(const float* __restrict__ wsf, const unsigned* __restrict__ wsu,
                        float* __restrict__ out) {
    if (threadIdx.x != 0 || blockIdx.x != 0) return;
    float ce = wsf[WS_CE] / (float)NPIX;
    float fo = wsf[WS_FOCAL] / (float)NPIX;
    float acc = 0.f, np = 0.f;
    for (int c = 0; c < NUM_CLASSES; ++c) {
        float cnt = (float)wsu[WS_COUNT + c];
        float uni = wsf[WS_PSUM + c] + cnt;
        if (cnt > 0.f) { np += 1.f; acc += (2.f * wsf[WS_INTER + c] + 1e-8f) / (uni + 1e-8f); }
    }
    float dice = (np > 0.f) ? (1.f - acc / np) : 1.f;
    float lov = 0.f;
    for (int c = 0; c < NUM_CLASSES; ++c) lov += wsf[WS_LOV + c];
    lov /= (float)NUM_CLASSES;
    out[0] = ce + dice + fo + 0.5f * lov;
}

extern "C" void kernel_launch(void* const* d_in, const int* in_sizes, int n_in,
                              void* d_out, int out_size, void* d_ws, size_t ws_size,
                              hipStream_t stream) {
    (void)in_sizes; (void)n_in; (void)out_size; (void)ws_size;
    const float* outputs = (const float*)d_in[0];
    const int*   masks   = (const int*)d_in[1];
    float*    wsf = (float*)d_ws;
    unsigned* wsu = (unsigned*)d_ws;

    k_zero<<<512, 256, 0, stream>>>(wsu);
    k_main<<<NBLK1, THR1, 0, stream>>>(outputs, masks, wsf, wsu);
    k_wmma_reduce<<<1, 32, 0, stream>>>(wsf + WS_PART, wsf + WS_PSUM);
    k_lovasz<<<NUM_CLASSES, 32, 0, stream>>>(wsu, wsf + WS_LOV);
    k_final<<<1, 1, 0, stream>>>(wsf, wsu, (float*)d_out);
}